// SecondaryCapsuleLayer_13039520711502
// MI455X (gfx1250) — compile-verified
//
#include <hip/hip_runtime.h>
#include <math.h>

// Problem constants (from reference): B=256, R=1152, C=10, O=16, I=8
#define B_SZ 256
#define R_SZ 1152
#define C_SZ 10
#define O_SZ 16
#define I_SZ 8
#define RI   (R_SZ * I_SZ)   // 9216, flattened (r,i) contraction dim
#define CO   (C_SZ * O_SZ)   // 160

typedef float v2f __attribute__((ext_vector_type(2)));
typedef float v8f __attribute__((ext_vector_type(8)));

// D = A(16x4 f32) * B(4x16 f32) + C(16x16 f32), wave32, full-precision f32 WMMA.
__device__ __forceinline__ v8f wmma_f32_16x16x4(v2f a, v2f b, v8f c) {
  return __builtin_amdgcn_wmma_f32_16x16x4_f32(false, a, false, b, (short)0, c,
                                               false, false);
}

__global__ void zero_f32_kernel(float* __restrict__ p, int n) {
  int i = blockIdx.x * blockDim.x + threadIdx.x;
  if (i < n) p[i] = 0.0f;
}

// Softmax over routes (axis=R) of b_logits[R,C] -> ccoef[R,C]. One block per c.
__global__ __launch_bounds__(128) void softmax_routes_kernel(
    const float* __restrict__ blog, float* __restrict__ ccoef) {
  const int c = blockIdx.x;
  const int t = threadIdx.x;
  __shared__ float sm[128];

  float mx = -INFINITY;
  for (int r = t; r < R_SZ; r += 128) mx = fmaxf(mx, blog[r * C_SZ + c]);
  sm[t] = mx;
  __syncthreads();
  for (int s = 64; s > 0; s >>= 1) {
    if (t < s) sm[t] = fmaxf(sm[t], sm[t + s]);
    __syncthreads();
  }
  mx = sm[0];
  __syncthreads();

  float sum = 0.0f;
  for (int r = t; r < R_SZ; r += 128) sum += __expf(blog[r * C_SZ + c] - mx);
  sm[t] = sum;
  __syncthreads();
  for (int s = 64; s > 0; s >>= 1) {
    if (t < s) sm[t] += sm[t + s];
    __syncthreads();
  }
  const float inv = 1.0f / sm[0];
  for (int r = t; r < R_SZ; r += 128)
    ccoef[r * C_SZ + c] = __expf(blog[r * C_SZ + c] - mx) * inv;
}

// Fused s + squash:  v[b,c,o] = squash( sum_{r,i} x[b,r,i] * ccoef[r,c]*W[r,c,o,i] )
// GEMM per capsule c: M=batch tile(16), N=o(16), K=(r,i)=9216, split over 4 waves.
__global__ __launch_bounds__(128) void s_squash_kernel(
    const float* __restrict__ x, const float* __restrict__ W,
    const float* __restrict__ ccoef, float* __restrict__ v) {
  const int bt   = blockIdx.x;        // batch tile: 16 tiles of 16
  const int c    = blockIdx.y;        // capsule
  const int wave = threadIdx.x >> 5;
  const int lane = threadIdx.x & 31;
  const int n    = lane & 15;         // A: batch-in-tile m; B: output channel o
  const int h    = lane >> 4;         // K-half selector (k = 2h + vgpr)
  const int b0   = bt * 16;

  v8f acc = {};
  const int r0 = wave * (R_SZ / 4);                        // 288 routes per wave
  const float* xrow = x + (b0 + n) * RI + 2 * h;           // + r*8 + i0 later
  for (int r = r0; r < r0 + (R_SZ / 4); ++r) {
    const float coef = ccoef[r * C_SZ + c];
    const float* xp = xrow + r * I_SZ;
    const float* wp = W + ((r * C_SZ + c) * O_SZ + n) * I_SZ + 2 * h;
#pragma unroll
    for (int i0 = 0; i0 < I_SZ; i0 += 4) {
      v2f a = *(const v2f*)(xp + i0);                      // A[m, k]=x[b0+m, r, i]
      v2f b = *(const v2f*)(wp + i0);                      // B[k, o]=W[r,c,o,i]
      b.x *= coef;
      b.y *= coef;
      acc = wmma_f32_16x16x4(a, b, acc);
    }
  }

  __shared__ v8f red[4][32];
  red[wave][lane] = acc;
  __syncthreads();
  if (wave == 0) {
    v8f s = red[0][lane];
#pragma unroll
    for (int w2 = 1; w2 < 4; ++w2) s += red[w2][lane];
    // D layout: VGPR j -> M = j + 8h, N = n
#pragma unroll
    for (int j = 0; j < 8; ++j) {
      const float sv = s[j];
      const float vv = sv * fabsf(sv) / (1.0f + sv * sv);  // squash (elementwise)
      v[(b0 + j + 8 * h) * CO + c * O_SZ + n] = vv;
    }
  }
}

// Agreement update: blog[r,c] += (1/B) * sum_{i,o} W[r,c,o,i] * G[ri,o],
// G[ri,o] = sum_b x[b,ri] * v[b,c,o]  -- computed as WMMA GEMM, G kept in regs.
// One block per (r-pair, c): M = 16 ri values (2 routes), N = 16 o, K = 256 b.
__global__ __launch_bounds__(128) void agreement_kernel(
    const float* __restrict__ x, const float* __restrict__ W,
    const float* __restrict__ v, float* __restrict__ blog) {
  const int rp   = blockIdx.x;        // route pair: 576 of them
  const int c    = blockIdx.y;
  const int wave = threadIdx.x >> 5;
  const int lane = threadIdx.x & 31;
  const int n    = lane & 15;         // A: ri-in-tile m; B: output channel o
  const int h    = lane >> 4;
  const int ri0  = rp * 16;

  v8f acc = {};
  const int kbase = wave * (B_SZ / 4);                     // 64 batches per wave
  for (int kb = 0; kb < (B_SZ / 4); kb += 4) {
    const int b0 = kbase + kb + 2 * h;                     // k = 2h + vgpr
    v2f a, b;
    a.x = x[b0 * RI + ri0 + n];                            // A[m,k] = x[b, ri0+m]
    a.y = x[(b0 + 1) * RI + ri0 + n];
    b.x = v[b0 * CO + c * O_SZ + n];                       // B[k,o] = v[b, c, o]
    b.y = v[(b0 + 1) * CO + c * O_SZ + n];
    acc = wmma_f32_16x16x4(a, b, acc);
  }

  __shared__ v8f red[4][32];
  red[wave][lane] = acc;
  __syncthreads();
  if (wave == 0) {
    v8f d = red[0][lane];
#pragma unroll
    for (int w2 = 1; w2 < 4; ++w2) d += red[w2][lane];
    // D[j + 8h, n] = G[ri0 + j + 8h, o=n]; route r = 2*rp + h, i = j.
    const int r = rp * 2 + h;
    const float* wp = W + ((r * C_SZ + c) * O_SZ + n) * I_SZ;
    float part = 0.0f;
#pragma unroll
    for (int j = 0; j < 8; ++j) part += d[j] * wp[j];
    // Reduce over o (16 lanes within each half; halves are independent routes).
#pragma unroll
    for (int mask = 1; mask < 16; mask <<= 1)
      part += __shfl_xor(part, mask, 32);
    if (n == 0)                       // unique writer per (r,c): no atomics needed
      blog[r * C_SZ + c] += part * (1.0f / (float)B_SZ);
  }
}

extern "C" void kernel_launch(void* const* d_in, const int* in_sizes, int n_in,
                              void* d_out, int out_size, void* d_ws, size_t ws_size,
                              hipStream_t stream) {
  const float* x = (const float*)d_in[0];   // [B, R, I]
  const float* W = (const float*)d_in[1];   // [R, C, O, I]
  float* v_out   = (float*)d_out;           // [B, C, O, 1] == [B, C, O] flat

  float* blog  = (float*)d_ws;              // [R, C] routing logits
  float* ccoef = blog + R_SZ * C_SZ;        // [R, C] softmax coefficients

  zero_f32_kernel<<<(R_SZ * C_SZ + 255) / 256, 256, 0, stream>>>(blog, R_SZ * C_SZ);

  for (int it = 0; it < 3; ++it) {
    softmax_routes_kernel<<<C_SZ, 128, 0, stream>>>(blog, ccoef);
    s_squash_kernel<<<dim3(B_SZ / 16, C_SZ), 128, 0, stream>>>(x, W, ccoef, v_out);
    if (it < 2)
      agreement_kernel<<<dim3(R_SZ / 2, C_SZ), 128, 0, stream>>>(x, W, v_out, blog);
  }
}